// AGNNNet_6356551598697
// MI455X (gfx1250) — compile-verified
//
#include <hip/hip_runtime.h>
#include <math.h>

typedef __attribute__((ext_vector_type(2))) float v2f;
typedef __attribute__((ext_vector_type(8))) float v8f;

#define FDIM 128   // input feature dim
#define HDIM 64    // hidden dim

// ---------------------------------------------------------------------------
// GEMM: h[N,64] = x[N,128] @ W1[128,64] + b1[64], via V_WMMA_F32_16X16X4_F32.
// One wave per 16x16 output tile; K loop of 128 in steps of 4.
// A-layout (16x4 f32): lanes 0-15 -> K=k+0/k+1 in a[0]/a[1], lanes 16-31 -> K=k+2/k+3.
// B-layout (4x16 f32): mirrored (col = lane&15).
// C/D layout (16x16 f32, 8 VGPRs): col = lane&15, row = vgpr + (lane>>4)*8.
// ---------------------------------------------------------------------------
__global__ __launch_bounds__(256) void gemm_wmma(
    const float* __restrict__ x, const float* __restrict__ W1,
    const float* __restrict__ b1, float* __restrict__ h, int N)
{
  int wave = (int)((blockIdx.x * blockDim.x + threadIdx.x) >> 5);
  int lane = threadIdx.x & 31;
  const int ntiles_n = HDIM / 16;          // 4
  int mt = wave / ntiles_n;
  int nt = wave - mt * ntiles_n;
  int m0 = mt * 16;
  if (m0 >= N) return;
  int n0 = nt * 16;
  int l15   = lane & 15;
  int khalf = (lane >> 4) << 1;            // 0 or 2

  // bias goes to every row of the tile; column = lane&15 in the C/D layout
  v8f acc;
  float bias = b1[n0 + l15];
#pragma unroll
  for (int i = 0; i < 8; ++i) acc[i] = bias;

  int rowA = m0 + l15; if (rowA > N - 1) rowA = N - 1;   // clamp (N%16==0 normally)
  const float* xrow = x + (size_t)rowA * FDIM;

#pragma unroll 4
  for (int k = 0; k < FDIM; k += 4) {
    v2f a, b;
    a[0] = xrow[k + khalf];
    a[1] = xrow[k + khalf + 1];
    b[0] = W1[(size_t)(k + khalf) * HDIM + n0 + l15];
    b[1] = W1[(size_t)(k + khalf + 1) * HDIM + n0 + l15];
    acc = __builtin_amdgcn_wmma_f32_16x16x4_f32(
        /*neg_a=*/false, a, /*neg_b=*/false, b,
        /*c_mod=*/(short)0, acc, /*reuse_a=*/false, /*reuse_b=*/false);
  }

  int rbase = m0 + ((lane >> 4) << 3);
  float* hp = h + (size_t)rbase * HDIM + n0 + l15;
  if (m0 + 16 <= N) {
    // full tile: uniform scalar branch, 8 straight stores (clause-able)
#pragma unroll
    for (int v = 0; v < 8; ++v) hp[(size_t)v * HDIM] = acc[v];
  } else {
    // boundary tile only (never taken when N % 16 == 0)
#pragma unroll
    for (int v = 0; v < 8; ++v)
      if (rbase + v < N) hp[(size_t)v * HDIM] = acc[v];
  }
}

// ---------------------------------------------------------------------------
// Per-node: xn = hin / max(||hin||,eps); also zero s[] and the next prop's
// output buffer. One wave32 per node row; each lane owns 2 of 64 features.
// ---------------------------------------------------------------------------
__global__ __launch_bounds__(256) void norm_zero(
    const float* __restrict__ hin, float* __restrict__ xn,
    float* __restrict__ s, float* __restrict__ hout, int N)
{
  int wave = (int)((blockIdx.x * blockDim.x + threadIdx.x) >> 5);
  int lane = threadIdx.x & 31;
  if (wave >= N) return;
  float2 v = ((const float2*)(hin + (size_t)wave * HDIM))[lane];
  float ss = v.x * v.x + v.y * v.y;
#pragma unroll
  for (int m = 16; m >= 1; m >>= 1) ss += __shfl_xor(ss, m, 32);
  float inv = 1.0f / fmaxf(sqrtf(ss), 1e-12f);
  float2 o; o.x = v.x * inv; o.y = v.y * inv;
  ((float2*)(xn + (size_t)wave * HDIM))[lane] = o;
  float2 z; z.x = 0.0f; z.y = 0.0f;
  ((float2*)(hout + (size_t)wave * HDIM))[lane] = z;
  if (lane == 0) s[wave] = 0.0f;
}

// ---------------------------------------------------------------------------
// Per-edge pass 1: w[e] = exp(beta*dot(xn[src],xn[dst]) - |beta|)  (constant
// shift |beta| replaces segment_max: softmax is shift-invariant and
// |beta*cos| <= |beta|).  atomicAdd into s[dst].  Edges e>=E are self loops.
// ---------------------------------------------------------------------------
__global__ __launch_bounds__(256) void edge_logits(
    const float* __restrict__ xn, const int* __restrict__ ei,
    float* __restrict__ w, float* __restrict__ s,
    const float* __restrict__ beta_ptr, int E, int N)
{
  long long wave = ((long long)blockIdx.x * blockDim.x + threadIdx.x) >> 5;
  int lane = threadIdx.x & 31;
  long long ET = (long long)E + N;
  if (wave >= ET) return;
  int src, dst;
  if (wave < E) { src = ei[wave]; dst = ei[(long long)E + wave]; }
  else          { src = dst = (int)(wave - E); }
  float2 a = ((const float2*)(xn + (size_t)src * HDIM))[lane];
  float2 b = ((const float2*)(xn + (size_t)dst * HDIM))[lane];
  float d = a.x * b.x + a.y * b.y;
#pragma unroll
  for (int m = 16; m >= 1; m >>= 1) d += __shfl_xor(d, m, 32);
  if (lane == 0) {
    float beta = beta_ptr ? beta_ptr[0] : 1.0f;
    float we = expf(beta * d - fabsf(beta));
    w[wave] = we;
    atomicAdd(&s[dst], we);          // global_atomic_add_f32
  }
}

// ---------------------------------------------------------------------------
// Per-edge pass 2: hout[dst] += (w[e]/s[dst]) * hin[src]
// ---------------------------------------------------------------------------
__global__ __launch_bounds__(256) void edge_scatter(
    const float* __restrict__ hin, const int* __restrict__ ei,
    const float* __restrict__ w, const float* __restrict__ s,
    float* __restrict__ hout, int E, int N)
{
  long long wave = ((long long)blockIdx.x * blockDim.x + threadIdx.x) >> 5;
  int lane = threadIdx.x & 31;
  long long ET = (long long)E + N;
  if (wave >= ET) return;
  int src, dst;
  if (wave < E) { src = ei[wave]; dst = ei[(long long)E + wave]; }
  else          { src = dst = (int)(wave - E); }
  float alpha = w[wave] / s[dst];
  float2 v = ((const float2*)(hin + (size_t)src * HDIM))[lane];
  float* o = hout + (size_t)dst * HDIM + (lane << 1);
  atomicAdd(o,     alpha * v.x);
  atomicAdd(o + 1, alpha * v.y);
}

// ---------------------------------------------------------------------------
// Row-wise log_softmax over H=64. One wave per row.
// ---------------------------------------------------------------------------
__global__ __launch_bounds__(256) void logsoftmax_rows(
    const float* __restrict__ hin, float* __restrict__ out, int N)
{
  int wave = (int)((blockIdx.x * blockDim.x + threadIdx.x) >> 5);
  int lane = threadIdx.x & 31;
  if (wave >= N) return;
  float2 v = ((const float2*)(hin + (size_t)wave * HDIM))[lane];
  float mx = fmaxf(v.x, v.y);
#pragma unroll
  for (int m = 16; m >= 1; m >>= 1) mx = fmaxf(mx, __shfl_xor(mx, m, 32));
  float es = expf(v.x - mx) + expf(v.y - mx);
#pragma unroll
  for (int m = 16; m >= 1; m >>= 1) es += __shfl_xor(es, m, 32);
  float lse = mx + logf(es);
  float2 o; o.x = v.x - lse; o.y = v.y - lse;
  ((float2*)(out + (size_t)wave * HDIM))[lane] = o;
}

// ---------------------------------------------------------------------------
extern "C" void kernel_launch(void* const* d_in, const int* in_sizes, int n_in,
                              void* d_out, int out_size, void* d_ws, size_t ws_size,
                              hipStream_t stream)
{
  const float* x     = (const float*)d_in[0];
  const int*   ei    = (const int*)  d_in[1];   // [2,E] flattened
  const float* W1    = (const float*)d_in[2];
  const float* b1    = (const float*)d_in[3];
  const float* beta2 = (const float*)d_in[4];

  int N = in_sizes[0] / FDIM;
  int E = in_sizes[1] / 2;

  float* ws = (float*)d_ws;
  float* h0 = ws;                                // [N,64] gemm out / prop2 out
  float* h1 = h0 + (size_t)N * HDIM;             // [N,64] prop1 out
  float* xn = h1 + (size_t)N * HDIM;             // [N,64] normalized rows
  float* s  = xn + (size_t)N * HDIM;             // [N]    softmax denominators
  float* w  = s + N;                             // [E+N]  edge weights

  dim3 blk(256);
  long long ET = (long long)E + N;
  int mtiles = (N + 15) / 16;
  dim3 g_gemm((unsigned)(((long long)mtiles * (HDIM / 16) * 32 + 255) / 256));
  dim3 g_node((unsigned)(((long long)N * 32 + 255) / 256));
  dim3 g_edge((unsigned)((ET * 32 + 255) / 256));

  // h0 = x @ W1 + b1   (WMMA)
  gemm_wmma<<<g_gemm, blk, 0, stream>>>(x, W1, b1, h0, N);

  // prop1 (beta = 1.0 fixed)
  norm_zero  <<<g_node, blk, 0, stream>>>(h0, xn, s, h1, N);
  edge_logits<<<g_edge, blk, 0, stream>>>(xn, ei, w, s, nullptr, E, N);
  edge_scatter<<<g_edge, blk, 0, stream>>>(h0, ei, w, s, h1, E, N);

  // prop2 (beta = beta2[0], read on device)
  norm_zero  <<<g_node, blk, 0, stream>>>(h1, xn, s, h0, N);
  edge_logits<<<g_edge, blk, 0, stream>>>(xn, ei, w, s, beta2, E, N);
  edge_scatter<<<g_edge, blk, 0, stream>>>(h1, ei, w, s, h0, E, N);

  // log_softmax -> d_out
  logsoftmax_rows<<<g_node, blk, 0, stream>>>(h0, (float*)d_out, N);
}